// _PhysicsVideoFallback_74174085202600
// MI455X (gfx1250) — compile-verified
//
#include <hip/hip_runtime.h>
#include <stdint.h>

// Problem constants (match reference)
#define Hh 1024
#define Ww 1024
#define NFRAMES 64
#define DTc    0.1f
#define ALPHAc 0.05f
#define VXc    0.1f
#define VYc    0.0f

typedef float v4f __attribute__((ext_vector_type(4)));

// One block per row (1024 blocks). 256 threads * 4 cells (float4) = 1024 cells.
// Rows y-1, y, y+1 are staged into LDS with gfx1250 async global->LDS copies.
__global__ __launch_bounds__(256)
void _pde_step_kernel(const float* __restrict__ src,
                      float* __restrict__ dst,
                      float* __restrict__ frame /* -> frames[t][0][0][0] */) {
    __shared__ float smem[3 * Ww];          // 12 KB of the 320 KB/WGP LDS

    const int y   = blockIdx.x;             // 0..1023
    const int tid = threadIdx.x;            // 0..255
    const int x0  = tid << 2;               // 0,4,...,1020 (16B aligned)
    const int ym  = (y + Hh - 1) & (Hh - 1);
    const int yp  = (y + 1) & (Hh - 1);

    // LDS byte offsets (low 32 bits of the generic pointer == offset from LDS base)
    uint32_t l0 = (uint32_t)(uintptr_t)(&smem[0 * Ww + x0]);
    uint32_t l1 = (uint32_t)(uintptr_t)(&smem[1 * Ww + x0]);
    uint32_t l2 = (uint32_t)(uintptr_t)(&smem[2 * Ww + x0]);
    const float* g0 = src + (size_t)ym * Ww + x0;
    const float* g1 = src + (size_t)y  * Ww + x0;
    const float* g2 = src + (size_t)yp * Ww + x0;

    // gfx1250 async tensor/memory path: global -> LDS, tracked by ASYNCcnt
    asm volatile("global_load_async_to_lds_b128 %0, %1, off"
                 :: "v"(l0), "v"(g0) : "memory");
    asm volatile("global_load_async_to_lds_b128 %0, %1, off"
                 :: "v"(l1), "v"(g1) : "memory");
    asm volatile("global_load_async_to_lds_b128 %0, %1, off"
                 :: "v"(l2), "v"(g2) : "memory");
    asm volatile("s_wait_asynccnt 0x0" ::: "memory");
    __syncthreads();   // all waves' async copies are complete past this point

    float un[4];
#pragma unroll
    for (int j = 0; j < 4; ++j) {
        const int x  = x0 + j;
        const int xl = (x + Ww - 1) & (Ww - 1);
        const int xr = (x + 1) & (Ww - 1);
        const float c  = smem[1 * Ww + x];
        const float lf = smem[1 * Ww + xl];
        const float rt = smem[1 * Ww + xr];
        const float up = smem[0 * Ww + x];    // row y-1  (roll +1 on axis -2)
        const float dn = smem[2 * Ww + x];    // row y+1
        const float lap   = lf + rt + up + dn - 4.0f * c;
        const float du_dx = c - lf;
        const float du_dy = c - up;
        float v = c + DTc * (ALPHAc * lap - VXc * du_dx - VYc * du_dy);
        v = fminf(fmaxf(v, 0.0f), 1.0f);
        un[j] = v;
    }

    v4f r4; r4.x = un[0];        r4.y = un[1];        r4.z = un[2];        r4.w = un[3];
    v4f g4; g4.x = 0.5f * un[0]; g4.y = 0.5f * un[1]; g4.z = 0.5f * un[2]; g4.w = 0.5f * un[3];
    v4f b4; b4.x = 1.0f - un[0]; b4.y = 1.0f - un[1]; b4.z = 1.0f - un[2]; b4.w = 1.0f - un[3];

    // State write: regular (RT) store so it stays resident in the 192MB L2
    // for the next step's async reads.
    *(v4f*)(dst + (size_t)y * Ww + x0) = r4;

    // Frame writes: 768MB total stream -> non-temporal so frames don't evict state.
    const size_t plane = (size_t)Hh * Ww;
    const size_t rowoff = (size_t)y * Ww + x0;
    __builtin_nontemporal_store(r4, (v4f*)(frame + 0 * plane + rowoff));
    __builtin_nontemporal_store(g4, (v4f*)(frame + 1 * plane + rowoff));
    __builtin_nontemporal_store(b4, (v4f*)(frame + 2 * plane + rowoff));
}

extern "C" void kernel_launch(void* const* d_in, const int* in_sizes, int n_in,
                              void* d_out, int out_size, void* d_ws, size_t ws_size,
                              hipStream_t stream) {
    const float* init = (const float*)d_in[0];   // (1, H, W) float32
    float* out = (float*)d_out;                  // (T, 3, H, W) float32
    float* ws  = (float*)d_ws;                   // need 2*H*W floats = 8 MB
    float* sA  = ws;
    float* sB  = ws + (size_t)Hh * Ww;

    for (int t = 0; t < NFRAMES; ++t) {
        const float* src = (t == 0) ? init : ((t & 1) ? sA : sB);
        float*       dst = (t & 1) ? sB : sA;
        float*     frame = out + (size_t)t * 3 * Hh * Ww;
        hipLaunchKernelGGL(_pde_step_kernel, dim3(Hh), dim3(256), 0, stream,
                           src, dst, frame);
    }
}